// SigmoidLoss_63007170232510
// MI455X (gfx1250) — compile-verified
//
#include <hip/hip_runtime.h>

typedef float v2f __attribute__((ext_vector_type(2)));
typedef float v8f __attribute__((ext_vector_type(8)));

#define SIDE 256
#define NROW 2048                 // (64/2) * 64 = b*e rows
#define BSTRIDE (256 * 64)        // elements per batch slice of embeddings

// Banded weight matrix, synthesized from the reference formula:
// W[s][t] = 1 - 0.05*d for d < 20 (d = t-s if t>=s else s-t-1), else 0.
__device__ __forceinline__ float wband(int s, int t) {
    int d = (t >= s) ? (t - s) : (s - t - 1);
    return (d < 20) ? (1.0f - 0.05f * (float)d) : 0.0f;
}

// ---------------------------------------------------------------------------
// Kernel 1: sum of softplus(f1[n,s]*f2[n,t]) over s,t for each row n.
// Trans-pipe bound. Raw v_exp_f32/v_log_f32 (base-2) with the ln2 scale
// hoisted out of the loop; two independent accumulator chains for dual-issue.
// ---------------------------------------------------------------------------
__global__ void sig_softplus_kernel(const float* __restrict__ E,
                                    float* __restrict__ partial1) {
    __shared__ float sh_f1[SIDE];
    __shared__ float red[256];
    const int n = blockIdx.x;
    const int t = threadIdx.x;
    const int b = n >> 6;
    const int e = n & 63;
    const float* E1 = E + (size_t)b * BSTRIDE + e;          // f1[n, s] = E1[s*64]
    const float* E2 = E + (size_t)(b + 32) * BSTRIDE + e;   // f2[n, t] = E2[t*64]

    sh_f1[t] = E1[(size_t)t * 64];
    const float x2 = E2[(size_t)t * 64];
    __syncthreads();

    const float LOG2E = 1.4426950408889634f;
    const float LN2   = 0.6931471805599453f;

    float acc_max = 0.0f;   // sum of max(z,0)
    float acc_l2  = 0.0f;   // sum of log2(1 + 2^(-|z|*log2e))
#pragma unroll 8
    for (int s = 0; s < SIDE; ++s) {
        float z = sh_f1[s] * x2;
        acc_max += fmaxf(z, 0.0f);
        // -|z| via free src modifiers on the multiply; v_exp_f32 is 2^x
        float ee = __builtin_amdgcn_exp2f(-__builtin_fabsf(z) * LOG2E);
        // 1+ee in (1,2]: v_log_f32 (log2) is well-conditioned here
        acc_l2 += __builtin_amdgcn_logf(1.0f + ee);
    }
    float acc = acc_max + LN2 * acc_l2;   // = sum softplus(z) over s for this t

    red[t] = acc;
    __syncthreads();
    for (int off = 128; off > 0; off >>= 1) {
        if (t < off) red[t] += red[t + off];
        __syncthreads();
    }
    if (t == 0) partial1[n] = red[0];
}

// ---------------------------------------------------------------------------
// Kernel 2: bilinear term  sum_n f1[n,:] * W * f2[n,:]^T  via f32 WMMA.
// One wave handles 16 n-rows. Per 16-wide t-tile, a transposed W tile is
// staged in LDS once, so the K-loop is pure ds_load_b64 + v_wmma:
//   D(16x16) += A(16x4 of f1) x B(4x16 of W),  K = s in steps of 4,
// then fused elementwise multiply with f2 and a wave32 reduction.
// ---------------------------------------------------------------------------
__global__ void sig_wterm_wmma_kernel(const float* __restrict__ E,
                                      float* __restrict__ partial2) {
    __shared__ float shF1[16][SIDE];   // A rows (f1)
    __shared__ float shF2[16][SIDE];   // f2 rows for the epilogue
    __shared__ float shWt[16][SIDE];   // staged W tile, transposed: shWt[t-t0][s]
    const int tid = threadIdx.x;       // 0..31, one wave32
    const int n0  = blockIdx.x * 16;

    for (int i = tid; i < 16 * SIDE; i += 32) {
        const int row = i >> 8;
        const int s   = i & 255;
        const int n   = n0 + row;
        const int b   = n >> 6;
        const int e   = n & 63;
        shF1[row][s] = E[(size_t)b * BSTRIDE + (size_t)s * 64 + e];
        shF2[row][s] = E[(size_t)(b + 32) * BSTRIDE + (size_t)s * 64 + e];
    }
    __syncthreads();

    const int half = tid >> 4;   // 0: lanes 0-15, 1: lanes 16-31
    const int l16  = tid & 15;

    float acc = 0.0f;
    for (int t0 = 0; t0 < SIDE; t0 += 16) {
        // Stage the 16x256 (transposed) W tile for this t-range.
        for (int i = tid; i < 16 * SIDE; i += 32) {
            const int tt = i >> 8;
            const int s  = i & 255;
            shWt[tt][s] = wband(s, t0 + tt);
        }
        __syncthreads();

        v8f c = {};
#pragma unroll 4
        for (int k = 0; k < 64; ++k) {
            const int sbase = k * 4 + half * 2;   // this lane-half's K pair (8B aligned)
            // A 16x4 f32: lanes 0-15 hold M=lane,(K0,K1); lanes 16-31 hold M,(K2,K3)
            v2f a  = *(const v2f*)&shF1[l16][sbase];
            // B 4x16 f32 (row striped across lanes): v0 = K0|K2, v1 = K1|K3
            v2f bf = *(const v2f*)&shWt[l16][sbase];
            c = __builtin_amdgcn_wmma_f32_16x16x4_f32(
                    /*neg_a=*/false, a, /*neg_b=*/false, bf,
                    /*c_mod=*/(short)0, c, /*reuse_a=*/false, /*reuse_b=*/false);
        }
        // C/D layout: lane l, vgpr r -> (M = r + (l>=16?8:0), N = l&15)
#pragma unroll
        for (int r = 0; r < 8; ++r) {
            const int M = r + half * 8;
            acc += c[r] * shF2[M][t0 + l16];
        }
        __syncthreads();   // before shWt is overwritten for the next tile
    }

    // wave32 butterfly reduction
    for (int m = 16; m >= 1; m >>= 1)
        acc += __shfl_xor(acc, m, 32);
    if (tid == 0) partial2[blockIdx.x] = acc;
}

// ---------------------------------------------------------------------------
// Kernel 3: final scalar = (sum(partial1) - sum(partial2)) / (256*256)
// ---------------------------------------------------------------------------
__global__ void sig_finalize_kernel(const float* __restrict__ p1,
                                    const float* __restrict__ p2,
                                    float* __restrict__ out) {
    __shared__ float red[256];
    const int t = threadIdx.x;
    float s = 0.0f;
    for (int i = t; i < NROW; i += 256) s += p1[i];
    for (int i = t; i < NROW / 16; i += 256) s -= p2[i];
    red[t] = s;
    __syncthreads();
    for (int off = 128; off > 0; off >>= 1) {
        if (t < off) red[t] += red[t + off];
        __syncthreads();
    }
    if (t == 0) out[0] = red[0] * (1.0f / 65536.0f);
}

extern "C" void kernel_launch(void* const* d_in, const int* in_sizes, int n_in,
                              void* d_out, int out_size, void* d_ws, size_t ws_size,
                              hipStream_t stream) {
    const float* E = (const float*)d_in[0];   // embeddings (64,256,64) f32
    float* p1 = (float*)d_ws;                 // 2048 row partials (softplus)
    float* p2 = p1 + NROW;                    // 128 block partials (W bilinear)

    sig_softplus_kernel<<<NROW, 256, 0, stream>>>(E, p1);
    sig_wterm_wmma_kernel<<<NROW / 16, 32, 0, stream>>>(E, p2);
    sig_finalize_kernel<<<1, 256, 0, stream>>>(p1, p2, (float*)d_out);
}